// CustomConvolutionLayer_54468775248359
// MI455X (gfx1250) — compile-verified
//
#include <hip/hip_runtime.h>
#include <hip/hip_bf16.h>

typedef float v2f __attribute__((ext_vector_type(2)));
typedef float v8f __attribute__((ext_vector_type(8)));

#define IH 512
#define IW 512
#define TILE_R 32          // output rows per block
#define TILE_C 64          // output cols per block
#define HALO 3
#define LDS_R (TILE_R + 6) // 38
#define LDS_C 72           // 70 used (+2 pad, also staged from x)

__global__ __launch_bounds__(256)
void conv7x7_wmma_f32(const float* __restrict__ x,
                      const float* __restrict__ w,
                      float* __restrict__ out) {
  __shared__ float xs[LDS_R][LDS_C];
  // Zero-padded Toeplitz B pairs: wp[kh][j] = ( w[kh][j-16], w[kh][j-15] ),
  // zero outside tap range [0,7). Indexed as j = 16 + d0, d0 in [-15, 22].
  __shared__ v2f wp[7][40];

  const int tid = threadIdx.x;
  const int b   = blockIdx.z;
  const int oh0 = blockIdx.y * TILE_R;
  const int ow0 = blockIdx.x * TILE_C;

  // Build the padded weight-pair table (once per block, 280 entries).
  for (int idx = tid; idx < 7 * 40; idx += 256) {
    int kh = idx / 40;
    int j  = idx - kh * 40;
    int d  = j - 16;
    v2f p;
    p.x = ((unsigned)d       < 7u) ? w[kh * 7 + d]     : 0.0f;
    p.y = ((unsigned)(d + 1) < 7u) ? w[kh * 7 + d + 1] : 0.0f;
    wp[kh][j] = p;
  }

  // Stage halo tile into LDS (zero outside the image; pad cols also staged from x).
  const float* xb = x + (size_t)b * IH * IW;
  for (int idx = tid; idx < LDS_R * LDS_C; idx += 256) {
    int r  = idx / LDS_C;
    int c  = idx - r * LDS_C;
    int gr = oh0 + r - HALO;
    int gc = ow0 + c - HALO;
    float v = 0.0f;
    if ((unsigned)gr < (unsigned)IH && (unsigned)gc < (unsigned)IW)
      v = xb[(size_t)gr * IW + gc];
    xs[r][c] = v;
  }
  __syncthreads();

  // 8 waves: 2 (row) x 4 (col) grid of 16x16 output sub-tiles.
  const int wave = tid >> 5;
  const int lane = tid & 31;
  const int r0   = (wave >> 2) * 16;   // 0 or 16
  const int c0   = (wave & 3) * 16;    // 0,16,32,48
  const int n    = lane & 15;          // M-row for A, N-col for B/C/D
  const int klo  = (lane >> 4) * 2;    // K-pair: lanes 0-15 -> (0,1), 16-31 -> (2,3)

  v8f acc = {};
#pragma unroll
  for (int kh = 0; kh < 7; ++kh) {
    // A row m needs x row (oh0 + r0 + m + kh - 3) -> LDS row r0 + m + kh.
    const float* arow = &xs[r0 + n + kh][c0 + klo];
    // B pair for chunk: wp[kh][16 + chunk*4 + klo - n]  (base index >= 1).
    const v2f* brow = &wp[kh][16 + klo - n];
#pragma unroll
    for (int chunk = 0; chunk < 6; ++chunk) {
      v2f a  = *(const v2f*)(arow + chunk * 4);  // ds_load_b64, imm offset
      v2f bv = brow[chunk * 4];                  // ds_load_b64, imm offset
      acc = __builtin_amdgcn_wmma_f32_16x16x4_f32(false, a, false, bv,
                                                  (short)0, acc, false, false);
    }
  }

  // C/D layout: VGPR v holds row v (lanes 0-15) and row v+8 (lanes 16-31),
  // column = lane&15 -> 64B coalesced segments per half-wave per row.
  const int half8 = (lane >> 4) * 8;
  float* op = out + ((size_t)b * IH + (oh0 + r0 + half8)) * IW + (ow0 + c0 + n);
#pragma unroll
  for (int v = 0; v < 8; ++v)
    op[(size_t)v * IW] = acc[v];
}

extern "C" void kernel_launch(void* const* d_in, const int* in_sizes, int n_in,
                              void* d_out, int out_size, void* d_ws, size_t ws_size,
                              hipStream_t stream) {
  const float* x = (const float*)d_in[0];
  const float* w = (const float*)d_in[1];
  float* out     = (float*)d_out;
  const int nbatch = in_sizes[0] / (IH * IW);   // 128
  dim3 grid(IW / TILE_C, IH / TILE_R, nbatch);  // (8, 16, 128)
  conv7x7_wmma_f32<<<grid, 256, 0, stream>>>(x, w, out);
}